// InteractionModule_72009421685009
// MI455X (gfx1250) — compile-verified
//
#include <hip/hip_runtime.h>

typedef float v2f __attribute__((ext_vector_type(2)));
typedef float v8f __attribute__((ext_vector_type(8)));

#define SO3_EPS 1e-6f

__device__ __forceinline__ float wave_sum32(float x) {
#pragma unroll
  for (int o = 16; o >= 1; o >>= 1) x += __shfl_xor(x, o, 32);
  return x;
}
// reduce within each 16-lane half (keeps halves separate: masks < 16)
__device__ __forceinline__ float half_sum16(float x) {
#pragma unroll
  for (int o = 8; o >= 1; o >>= 1) x += __shfl_xor(x, o, 32);
  return x;
}
__device__ __forceinline__ unsigned fenc(float f) {
  unsigned u = __float_as_uint(f);
  return (u & 0x80000000u) ? ~u : (u | 0x80000000u);
}
__device__ __forceinline__ float fdec(unsigned u) {
  return (u & 0x80000000u) ? __uint_as_float(u ^ 0x80000000u)
                           : __uint_as_float(~u);
}
__device__ __forceinline__ float silu_f(float x) { return x / (1.f + __expf(-x)); }

// ---------------------------------------------------------------------------
// Pack W[K x N] (row major) into paired-row layout:
//   Wp[kp][2*col + j] = W[2*kp + j][col],  j in {0,1}
// so a WMMA B fragment {W[k+2h][col], W[k+2h+1][col]} is ONE b64 load.
// ---------------------------------------------------------------------------
__global__ void pack_pairs_k(const float* __restrict__ W, float* __restrict__ Wp,
                             int N) {
  const int kp = blockIdx.x;
  const int col = blockIdx.y * blockDim.x + threadIdx.x;
  if (col < N) {
    const float lo = W[(size_t)(2 * kp) * N + col];
    const float hi = W[(size_t)(2 * kp + 1) * N + col];
    float* dst = Wp + (size_t)kp * 2 * N + 2 * col;
    dst[0] = lo;
    dst[1] = hi;
  }
}

// ---------------------------------------------------------------------------
// fp32 WMMA GEMM: C[rows x N] = act(A[rows x K] @ W[K x N] + bias)
// W supplied PACKED (see pack_pairs_k). Wave tile: (16*RT) rows x (16*NT)
// cols, 4 waves/block. Per k-step: RT b64 A-loads + NT b64 B-loads feed
// RT*NT v_wmma_f32_16x16x4_f32.
// ---------------------------------------------------------------------------
template <int NT, int RT, int ACT>
__global__ __launch_bounds__(128) void gemm_f32_wmma(
    const float* __restrict__ A, const float* __restrict__ Wp,
    const float* __restrict__ bias, float* __restrict__ C, int K, int N) {
  const int lane = threadIdx.x & 31;
  const int wid = threadIdx.x >> 5;
  const int half = lane >> 4;
  const int r = lane & 15;
  const int row0 = (blockIdx.x * 4 + wid) * (16 * RT);
  const int col0 = blockIdx.y * (16 * NT);

  v8f acc[RT][NT] = {};
  const float* Ap[RT];
#pragma unroll
  for (int rt = 0; rt < RT; ++rt)
    Ap[rt] = A + (size_t)(row0 + 16 * rt + r) * K + 2 * half;
  const float* Bbase = Wp + (size_t)half * 2 * N + 2 * (col0 + r);

  for (int k0 = 0; k0 < K; k0 += 4) {
    v2f a[RT];
#pragma unroll
    for (int rt = 0; rt < RT; ++rt) a[rt] = *(const v2f*)(Ap[rt] + k0);
    const float* Bp = Bbase + (size_t)(k0 >> 1) * 2 * N;
#pragma unroll
    for (int i = 0; i < NT; ++i) {
      v2f b = *(const v2f*)(Bp + 32 * i);
#pragma unroll
      for (int rt = 0; rt < RT; ++rt)
        acc[rt][i] = __builtin_amdgcn_wmma_f32_16x16x4_f32(
            false, a[rt], false, b, (short)0, acc[rt][i], false, false);
    }
  }
#pragma unroll
  for (int rt = 0; rt < RT; ++rt) {
#pragma unroll
    for (int i = 0; i < NT; ++i) {
      const int col = col0 + 16 * i + r;
      const float bs = bias ? bias[col] : 0.f;
#pragma unroll
      for (int v = 0; v < 8; ++v) {
        const int row = row0 + 16 * rt + half * 8 + v;
        float x = acc[rt][i][v] + bs;
        if (ACT == 1) x = silu_f(x);
        C[(size_t)row * N + col] = x;
      }
    }
  }
}

// ---------------------------------------------------------------------------
// Q GEMM fused with attention logits: Q = EFA @ Wq (packed, N=512, no bias),
// logits[e,m] = 4 * sum_d Q[e,16m+d]*Kmat[e,16m+d]; atomicMax per (node,m).
// Wave tile: 32 rows x 64 cols (RT=2, NT=4).
// ---------------------------------------------------------------------------
__global__ __launch_bounds__(128) void gemm_qk_logits(
    const float* __restrict__ A, const float* __restrict__ Wp,
    const float* __restrict__ Kmat, const int* __restrict__ ec,
    float* __restrict__ logits, unsigned* __restrict__ nmax, int K) {
  const int lane = threadIdx.x & 31;
  const int wid = threadIdx.x >> 5;
  const int half = lane >> 4;
  const int r = lane & 15;
  const int row0 = (blockIdx.x * 4 + wid) * 32;
  const int col0 = blockIdx.y * 64;

  v8f acc[2][4] = {};
  const float* Ap[2];
#pragma unroll
  for (int rt = 0; rt < 2; ++rt)
    Ap[rt] = A + (size_t)(row0 + 16 * rt + r) * K + 2 * half;
  const float* Bbase = Wp + (size_t)half * 1024 + 2 * (col0 + r);

  for (int k0 = 0; k0 < K; k0 += 4) {
    v2f a[2];
#pragma unroll
    for (int rt = 0; rt < 2; ++rt) a[rt] = *(const v2f*)(Ap[rt] + k0);
    const float* Bp = Bbase + (size_t)(k0 >> 1) * 1024;
#pragma unroll
    for (int i = 0; i < 4; ++i) {
      v2f b = *(const v2f*)(Bp + 32 * i);
#pragma unroll
      for (int rt = 0; rt < 2; ++rt)
        acc[rt][i] = __builtin_amdgcn_wmma_f32_16x16x4_f32(
            false, a[rt], false, b, (short)0, acc[rt][i], false, false);
    }
  }
#pragma unroll
  for (int rt = 0; rt < 2; ++rt) {
#pragma unroll
    for (int i = 0; i < 4; ++i) {
      const int col = col0 + 16 * i + r;
      const int head = (col0 >> 4) + i;
      float sums[8];
#pragma unroll
      for (int v = 0; v < 8; ++v) {
        const int row = row0 + 16 * rt + half * 8 + v;
        float t = acc[rt][i][v] * Kmat[(size_t)row * 512 + col];
        sums[v] = half_sum16(t) * 4.0f;  // * ISQRTD
      }
      if (r == 0) {
#pragma unroll
        for (int v = 0; v < 8; ++v) {
          const int row = row0 + 16 * rt + half * 8 + v;
          logits[(size_t)row * 32 + head] = sums[v];
          atomicMax(&nmax[(size_t)ec[row] * 32 + head], fenc(sums[v]));
        }
      }
    }
  }
}

// ---------------------------------------------------------------------------
// LayerNorm over 256 features + cutoff scaling of first 128 (in place).
// ---------------------------------------------------------------------------
__global__ __launch_bounds__(256) void layernorm_cutoff_k(
    float* __restrict__ lat, const float* __restrict__ ln_g,
    const float* __restrict__ ln_b, const float* __restrict__ cutoff) {
  const int wid = threadIdx.x >> 5, lane = threadIdx.x & 31;
  const int e = blockIdx.x * 8 + wid;
  float* row = lat + (size_t)e * 256;
  float x[8], s = 0.f, sq = 0.f;
#pragma unroll
  for (int j = 0; j < 8; ++j) {
    x[j] = row[lane + 32 * j];
    s += x[j];
    sq += x[j] * x[j];
  }
  s = wave_sum32(s);
  sq = wave_sum32(sq);
  const float mu = s * (1.f / 256.f);
  const float var = sq * (1.f / 256.f) - mu * mu;
  const float inv = rsqrtf(var + 1e-5f);
  const float cc = cutoff[e];
#pragma unroll
  for (int j = 0; j < 8; ++j) {
    const int c = lane + 32 * j;
    float y = (x[j] - mu) * inv * ln_g[c] + ln_b[c];
    if (c < 128) y *= cc;
    row[c] = y;
  }
}

// efa[e] = [node_inv[ec[e]] | node_inv[en[e]] | edge_inv[e]]  (192 cols)
__global__ void efa_gather_k(const float* __restrict__ ninv,
                             const float* __restrict__ einv,
                             const int* __restrict__ ec,
                             const int* __restrict__ en,
                             float* __restrict__ efa) {
  const int e = blockIdx.x, j = threadIdx.x;
  float v;
  if (j < 64)
    v = ninv[(size_t)ec[e] * 64 + j];
  else if (j < 128)
    v = ninv[(size_t)en[e] * 64 + (j - 64)];
  else
    v = einv[(size_t)e * 64 + (j - 128)];
  efa[(size_t)e * 192 + j] = v;
}

// ex = exp(logit - segment_max); accumulate segment sums. (in place on logits)
__global__ void softmax_exp_k(float* __restrict__ ex,
                              const unsigned* __restrict__ nmax,
                              float* __restrict__ nsum,
                              const int* __restrict__ ec) {
  const int idx = blockIdx.x * blockDim.x + threadIdx.x;
  const int e = idx >> 5, m = idx & 31;
  const float mx = fdec(nmax[(size_t)ec[e] * 32 + m]);
  const float v = __expf(ex[idx] - mx);
  ex[idx] = v;
  atomicAdd(&nsum[(size_t)ec[e] * 32 + m], v);
}

// emb = weight_channels(edge_attr, w_env) * mul * att; segment-sum into env.
__global__ void env_accum_k(const float* __restrict__ ex,
                            const float* __restrict__ nsum,
                            const float* __restrict__ wts,
                            const float* __restrict__ eattr,
                            const int* __restrict__ ec,
                            float* __restrict__ env) {
  const int idx = blockIdx.x * blockDim.x + threadIdx.x;
  const int e = idx >> 5, m = idx & 31;
  const int node = ec[e];
  const float att = ex[idx] / (nsum[(size_t)node * 32 + m] + 1e-16f);
  const float* wr = wts + (size_t)e * 160;
  const float w0 = wr[64 + 2 * m], w1 = wr[64 + 2 * m + 1], g = wr[128 + m];
  const float4 a = *(const float4*)(eattr + (size_t)e * 128 + 4 * m);
  float* dst = env + (size_t)node * 128 + 4 * m;
  atomicAdd(dst + 0, a.x * w0 * g * att);
  atomicAdd(dst + 1, a.y * w1 * g * att);
  atomicAdd(dst + 2, a.z * w1 * g * att);
  atomicAdd(dst + 3, a.w * w1 * g * att);
}

// Per node: SO(3) norm of env then per-channel 32x32 mixes (env_lin_w).
__global__ __launch_bounds__(256) void node_env_k(
    const float* __restrict__ env, const float* __restrict__ env_norm_w,
    const float* __restrict__ env_lin_w, float* __restrict__ env2) {
  __shared__ float wl[2][32][32];   // 8 KB
  __shared__ float nd[8][32][4];    // 4 KB
  for (int i = threadIdx.x; i < 2048; i += 256) ((float*)wl)[i] = env_lin_w[i];
  const int wid = threadIdx.x >> 5, m = threadIdx.x & 31;
  const int n = blockIdx.x * 8 + wid;
  const float4 x = *(const float4*)(env + (size_t)n * 128 + 4 * m);
  const float ssum = wave_sum32(x.x * x.x);
  const float vsum = wave_sum32(x.y * x.y + x.z * x.z + x.w * x.w);
  const float sr = rsqrtf(ssum * (1.f / 32.f) + SO3_EPS) * env_norm_w[2 * m];
  const float vr = rsqrtf(vsum * (1.f / 96.f) + SO3_EPS) * env_norm_w[2 * m + 1];
  nd[wid][m][0] = x.x * sr;
  nd[wid][m][1] = x.y * vr;
  nd[wid][m][2] = x.z * vr;
  nd[wid][m][3] = x.w * vr;
  __syncthreads();
  float o0 = 0, o1 = 0, o2 = 0, o3 = 0;
#pragma unroll
  for (int mm = 0; mm < 32; ++mm) {
    const float w0 = wl[0][mm][m], w1 = wl[1][mm][m];
    o0 += nd[wid][mm][0] * w0;
    o1 += nd[wid][mm][1] * w1;
    o2 += nd[wid][mm][2] * w1;
    o3 += nd[wid][mm][3] * w1;
  }
  float4 o = {o0, o1, o2, o3};
  *(float4*)(env2 + (size_t)n * 128 + 4 * m) = o;
}

// Per edge: eq SO(3) norm, tensor product with env_e, TP norms, lin_vec,
// and final output assembly [latents(256) | scalars(64) | vec(96)].
__global__ __launch_bounds__(256) void final_edge_k(
    const float* __restrict__ lat, const float* __restrict__ wts,
    const float* __restrict__ eqf, const float* __restrict__ eq_norm_w,
    const float* __restrict__ env2, const float* __restrict__ tp_norm_w,
    const float* __restrict__ lin_vec_w, const int* __restrict__ ec,
    float* __restrict__ out) {
  __shared__ float wlv[96][32];       // 12 KB
  __shared__ float ovl[8][96][3];     // 9 KB
  for (int i = threadIdx.x; i < 96 * 32; i += 256)
    ((float*)wlv)[i] = lin_vec_w[i];
  const int wid = threadIdx.x >> 5, m = threadIdx.x & 31;
  const int e = blockIdx.x * 8 + wid;

  // eq = so3_norm(weight_channels(eq_features, w_eq), eq_norm_w)
  const float* wr = wts + (size_t)e * 160;
  const float w0 = wr[2 * m], w1 = wr[2 * m + 1];
  const float4 f = *(const float4*)(eqf + (size_t)e * 128 + 4 * m);
  float s = f.x * w0;
  float vx = f.y * w1, vy = f.z * w1, vz = f.w * w1;
  const float sm = wave_sum32(s * s) * (1.f / 32.f);
  const float vm = wave_sum32(vx * vx + vy * vy + vz * vz) * (1.f / 96.f);
  s *= rsqrtf(sm + SO3_EPS) * eq_norm_w[2 * m];
  const float vr = rsqrtf(vm + SO3_EPS) * eq_norm_w[2 * m + 1];
  vx *= vr; vy *= vr; vz *= vr;

  const int node = ec[e];
  const float4 en = *(const float4*)(env2 + (size_t)node * 128 + 4 * m);
  const float es = en.x, ex_ = en.y, ey = en.z, ez = en.w;

  float os0 = s * es;
  float os1 = (vx * ex_ + vy * ey + vz * ez) * 0.57735026918962576f; // /sqrt(3)
  float ov[3][3];
  ov[0][0] = s * ex_; ov[0][1] = s * ey; ov[0][2] = s * ez;
  ov[1][0] = vx * es; ov[1][1] = vy * es; ov[1][2] = vz * es;
  const float is2 = 0.70710678118654752f;
  ov[2][0] = (vy * ez - vz * ey) * is2;
  ov[2][1] = (vz * ex_ - vx * ez) * is2;
  ov[2][2] = (vx * ey - vy * ex_) * is2;

  const float m0 = wave_sum32(os0 * os0) * (1.f / 32.f);
  const float m1 = wave_sum32(os1 * os1) * (1.f / 32.f);
  os0 *= rsqrtf(m0 + SO3_EPS) * tp_norm_w[5 * m + 0];
  os1 *= rsqrtf(m1 + SO3_EPS) * tp_norm_w[5 * m + 1];
#pragma unroll
  for (int t = 0; t < 3; ++t) {
    const float mt =
        wave_sum32(ov[t][0] * ov[t][0] + ov[t][1] * ov[t][1] +
                   ov[t][2] * ov[t][2]) * (1.f / 96.f);
    const float sc = rsqrtf(mt + SO3_EPS) * tp_norm_w[5 * m + 2 + t];
    ovl[wid][3 * m + t][0] = ov[t][0] * sc;
    ovl[wid][3 * m + t][1] = ov[t][1] * sc;
    ovl[wid][3 * m + t][2] = ov[t][2] * sc;
  }
  __syncthreads();

  float* orow = out + (size_t)e * 416;
#pragma unroll
  for (int j = 0; j < 8; ++j)
    orow[m + 32 * j] = lat[(size_t)e * 256 + m + 32 * j];
  orow[256 + 2 * m] = os0;
  orow[256 + 2 * m + 1] = os1;

  float v0 = 0, v1 = 0, v2 = 0;  // lane m == output channel o
#pragma unroll 4
  for (int i = 0; i < 96; ++i) {
    const float w = wlv[i][m];
    v0 += ovl[wid][i][0] * w;
    v1 += ovl[wid][i][1] * w;
    v2 += ovl[wid][i][2] * w;
  }
  orow[320 + 3 * m + 0] = v0;
  orow[320 + 3 * m + 1] = v1;
  orow[320 + 3 * m + 2] = v2;
}

// ---------------------------------------------------------------------------
extern "C" void kernel_launch(void* const* d_in, const int* in_sizes, int n_in,
                              void* d_out, int out_size, void* d_ws,
                              size_t ws_size, hipStream_t stream) {
  const float* X      = (const float*)d_in[0];
  const float* cutoff = (const float*)d_in[1];
  const float* eqf    = (const float*)d_in[2];
  const float* eattr  = (const float*)d_in[3];
  const float* ninv   = (const float*)d_in[4];
  const float* einv   = (const float*)d_in[5];
  const float* W1  = (const float*)d_in[6];
  const float* b1  = (const float*)d_in[7];
  const float* W2  = (const float*)d_in[8];
  const float* b2  = (const float*)d_in[9];
  const float* lng = (const float*)d_in[10];
  const float* lnb = (const float*)d_in[11];
  const float* We1 = (const float*)d_in[12];
  const float* be1 = (const float*)d_in[13];
  const float* We2 = (const float*)d_in[14];
  const float* be2 = (const float*)d_in[15];
  const float* Wq  = (const float*)d_in[16];
  const float* Wk  = (const float*)d_in[17];
  const float* eq_norm_w  = (const float*)d_in[18];
  const float* env_norm_w = (const float*)d_in[19];
  const float* env_lin_w  = (const float*)d_in[20];
  const float* tp_norm_w  = (const float*)d_in[21];
  const float* lin_vec_w  = (const float*)d_in[22];
  const int* ec = (const int*)d_in[23];
  const int* en = (const int*)d_in[24];
  (void)n_in; (void)out_size; (void)ws_size;

  const int E = in_sizes[1];           // cutoff_coeffs has E elements
  const int Nn = in_sizes[4] / 64;     // node_invariants: [N, 64]
  const size_t Es = (size_t)E;

  float* ws   = (float*)d_ws;
  float* lat  = ws;                    // [E,256]
  float* h    = ws + Es * 256;         // [E,256] h1 / he1 / efa(192)
  float* kmat = ws + Es * 512;         // [E,512]
  float* wts  = ws + Es * 1024;        // [E,160]
  float* exb  = ws + Es * 1184;        // [E,32] logits -> exp
  unsigned* nmax = (unsigned*)(ws + Es * 1216);        // [N,32]
  float* nsum = ws + Es * 1216 + (size_t)Nn * 32;      // [N,32]
  float* env  = ws + Es * 1216 + (size_t)Nn * 64;      // [N,32,4]
  float* env2 = ws + Es * 1216 + (size_t)Nn * 192;     // [N,32,4]
  float* wpk0 = ws + Es * 1216 + (size_t)Nn * 320;     // packed weights:
  float* wp1  = wpk0;                  // 192x256 -> 49152
  float* wp2  = wp1 + 49152;           // 256x256 -> 65536
  float* wpe1 = wp2 + 65536;           // 256x256 -> 65536
  float* wpe2 = wpe1 + 65536;          // 256x160 -> 40960
  float* wpk  = wpe2 + 40960;          // 256x512 -> 131072
  float* wpq  = wpk + 131072;          // 192x512 -> 98304
  float* out = (float*)d_out;

  // pack the six weight matrices into paired-row layout (L2-resident)
  pack_pairs_k<<<dim3(96, 1), 256, 0, stream>>>(W1, wp1, 256);
  pack_pairs_k<<<dim3(128, 1), 256, 0, stream>>>(W2, wp2, 256);
  pack_pairs_k<<<dim3(128, 1), 256, 0, stream>>>(We1, wpe1, 256);
  pack_pairs_k<<<dim3(128, 1), 256, 0, stream>>>(We2, wpe2, 160);
  pack_pairs_k<<<dim3(128, 2), 256, 0, stream>>>(Wk, wpk, 512);
  pack_pairs_k<<<dim3(96, 2), 256, 0, stream>>>(Wq, wpq, 512);

  // zero nmax (ordered-uint 0 == -inf), nsum, env (contiguous region)
  hipMemsetAsync(nmax, 0, (size_t)Nn * 192 * sizeof(float), stream);

  const dim3 blk(128);
  // 1) h1 = silu(X @ W1 + b1)
  gemm_f32_wmma<4, 2, 1><<<dim3(E / 128, 4), blk, 0, stream>>>(X, wp1, b1, h, 192, 256);
  // 2) nl = h1 @ W2 + b2  (into lat, LN is done in place next)
  gemm_f32_wmma<4, 2, 0><<<dim3(E / 128, 4), blk, 0, stream>>>(h, wp2, b2, lat, 256, 256);
  // 3) layernorm + cutoff -> latents
  layernorm_cutoff_k<<<E / 8, 256, 0, stream>>>(lat, lng, lnb, cutoff);
  // 4) he1 = silu(lat @ We1 + be1)
  gemm_f32_wmma<4, 2, 1><<<dim3(E / 128, 4), blk, 0, stream>>>(lat, wpe1, be1, h, 256, 256);
  // 5) weights = he1 @ We2 + be2   (N=160 -> NT=2, grid.y=5)
  gemm_f32_wmma<2, 2, 0><<<dim3(E / 128, 5), blk, 0, stream>>>(h, wpe2, be2, wts, 256, 160);
  // 6) K = lat @ Wk
  gemm_f32_wmma<4, 2, 0><<<dim3(E / 128, 8), blk, 0, stream>>>(lat, wpk, nullptr, kmat, 256, 512);
  // 7) efa gather (reuses h; he1 is dead)
  efa_gather_k<<<E, 192, 0, stream>>>(ninv, einv, ec, en, h);
  // 8) Q GEMM fused with logits + segment max
  gemm_qk_logits<<<dim3(E / 128, 8), blk, 0, stream>>>(h, wpq, kmat, ec, exb, nmax, 192);
  // 9) exp + segment sum
  softmax_exp_k<<<(E * 32) / 256, 256, 0, stream>>>(exb, nmax, nsum, ec);
  // 10) att-weighted env accumulation
  env_accum_k<<<(E * 32) / 256, 256, 0, stream>>>(exb, nsum, wts, eattr, ec, env);
  // 11) node-level SO3 norm + channel mixing
  node_env_k<<<Nn / 8, 256, 0, stream>>>(env, env_norm_w, env_lin_w, env2);
  // 12) tensor product + output assembly
  final_edge_k<<<E / 8, 256, 0, stream>>>(lat, wts, eqf, eq_norm_w, env2,
                                          tp_norm_w, lin_vec_w, ec, out);
}